// GatedEncoderLayer_867583394131
// MI455X (gfx1250) — compile-verified
//
#include <hip/hip_runtime.h>
#include <hip/hip_bf16.h>

// Problem constants (match reference)
#define Bd  2
#define Sd  2048
#define Dd  512
#define Hh  8
#define FCd 2048
#define HD  64          // head dim
#define LN_EPS 1e-5f

typedef float v2f __attribute__((ext_vector_type(2)));
typedef float v8f __attribute__((ext_vector_type(8)));

__device__ __forceinline__ v8f wmma_f32(v2f a, v2f b, v8f c) {
  // D = A(16x4 f32) * B(4x16 f32) + C(16x16 f32)
  // 8 args: (neg_a, A, neg_b, B, c_mod, C, reuse_a, reuse_b)
  return __builtin_amdgcn_wmma_f32_16x16x4_f32(
      false, a, false, b, (short)0, c, false, false);
}

// -------- CDNA5 async global->LDS staging (guarded; falls back to sync) ----
#if __has_builtin(__builtin_amdgcn_global_load_async_to_lds_b128) && \
    __has_builtin(__builtin_amdgcn_s_wait_asynccnt)
#define USE_ASYNC_LDS 1
#else
#define USE_ASYNC_LDS 0
#endif

#if USE_ASYNC_LDS
// Builtin prototype (from hipcc diagnostic): first arg is a pointer to a
// 16-byte int vector in the global (__device__) address space.
typedef int v4i_b __attribute__((vector_size(16)));
typedef __attribute__((address_space(1))) v4i_b gv4i;
typedef __attribute__((address_space(3))) v4i_b lv4i;

__device__ __forceinline__ void async_copy16(const float* g, float* l) {
  // GLOBAL_LOAD_ASYNC_TO_LDS_B128: per-lane 16B, tracked by ASYNCcnt
  __builtin_amdgcn_global_load_async_to_lds_b128(
      (gv4i*)(float*)g, (lv4i*)l, /*imm offset*/ 0, /*cpol*/ 0);
}
#endif

// ---------------------------------------------------------------------------
// Generic f32 WMMA GEMM:  C[M,N] = A[M,K] * W[N,K]^T + bias[N]  (opt. SiLU)
// Block: 256 threads = 8 waves. Block tile 64x64, wave tile 16x32 (2 acc).
// K-tiles staged in LDS; double-buffered with async global->LDS when available.
// M,N multiples of 64; K multiple of 16. All true for this layer.
// ---------------------------------------------------------------------------
template <int SILU>
__global__ __launch_bounds__(256) void gemm_wmma_f32(
    const float* __restrict__ A, const float* __restrict__ W,
    const float* __restrict__ bias, float* __restrict__ C,
    int M, int N, int K) {
  __shared__ __align__(16) float Asl[2][64][16];
  __shared__ __align__(16) float Wsl[2][64][16];

  const int tid  = threadIdx.x;
  const int lane = tid & 31;
  const int wv   = tid >> 5;
  const int half = lane >> 4;
  const int l16  = lane & 15;
  const int mt   = wv & 3;          // wave m-tile 0..3
  const int nt0  = (wv >> 2) * 2;   // wave n-tiles {0,1} or {2,3}
  const int bm0  = blockIdx.y * 64;
  const int bn0  = blockIdx.x * 64;

  const int srow = tid >> 2;        // 0..63 staging row
  const int scol = (tid & 3) * 4;   // 0,4,8,12 staging col (float4)

  const float* Ag = A + (size_t)(bm0 + srow) * K + scol;
  const float* Wg = W + (size_t)(bn0 + srow) * K + scol;

  v8f acc0 = {}; v8f acc1 = {};

#if USE_ASYNC_LDS
  // prologue: stage first K-tile into buffer 0
  async_copy16(Ag, &Asl[0][srow][scol]);
  async_copy16(Wg, &Wsl[0][srow][scol]);
  int buf = 0;
  for (int k0 = 0; k0 < K; k0 += 16) {
    __builtin_amdgcn_s_wait_asynccnt(0);  // this wave's tile is in LDS
    __syncthreads();                      // all waves' tiles in LDS; prev reads done
    if (k0 + 16 < K) {                    // issue next tile into the other buffer
      async_copy16(Ag + k0 + 16, &Asl[buf ^ 1][srow][scol]);
      async_copy16(Wg + k0 + 16, &Wsl[buf ^ 1][srow][scol]);
    }
    const float(*Ab)[16] = Asl[buf];
    const float(*Wb)[16] = Wsl[buf];
#pragma unroll
    for (int t = 0; t < 4; ++t) {
      const int kk = 4 * t + 2 * half;
      v2f a, b0, b1;
      a.x  = Ab[mt * 16 + l16][kk];        a.y  = Ab[mt * 16 + l16][kk + 1];
      b0.x = Wb[nt0 * 16 + l16][kk];       b0.y = Wb[nt0 * 16 + l16][kk + 1];
      b1.x = Wb[(nt0 + 1) * 16 + l16][kk]; b1.y = Wb[(nt0 + 1) * 16 + l16][kk + 1];
      acc0 = wmma_f32(a, b0, acc0);
      acc1 = wmma_f32(a, b1, acc1);
    }
    buf ^= 1;
  }
#else
  for (int k0 = 0; k0 < K; k0 += 16) {
    const float4 av  = *(const float4*)(Ag + k0);
    const float4 wvv = *(const float4*)(Wg + k0);
    __syncthreads();  // previous iteration's readers done
    *(float4*)&Asl[0][srow][scol] = av;
    *(float4*)&Wsl[0][srow][scol] = wvv;
    __syncthreads();
#pragma unroll
    for (int t = 0; t < 4; ++t) {
      const int kk = 4 * t + 2 * half;
      v2f a, b0, b1;
      a.x  = Asl[0][mt * 16 + l16][kk];        a.y  = Asl[0][mt * 16 + l16][kk + 1];
      b0.x = Wsl[0][nt0 * 16 + l16][kk];       b0.y = Wsl[0][nt0 * 16 + l16][kk + 1];
      b1.x = Wsl[0][(nt0 + 1) * 16 + l16][kk]; b1.y = Wsl[0][(nt0 + 1) * 16 + l16][kk + 1];
      acc0 = wmma_f32(a, b0, acc0);
      acc1 = wmma_f32(a, b1, acc1);
    }
  }
#endif

#pragma unroll
  for (int r = 0; r < 8; ++r) {
    const int m = bm0 + mt * 16 + r + 8 * half;
    {
      const int n = bn0 + nt0 * 16 + l16;
      float v = acc0[r] + bias[n];
      if (SILU) v = v / (1.0f + expf(-v));
      C[(size_t)m * N + n] = v;
    }
    {
      const int n = bn0 + (nt0 + 1) * 16 + l16;
      float v = acc1[r] + bias[n];
      if (SILU) v = v / (1.0f + expf(-v));
      C[(size_t)m * N + n] = v;
    }
  }
}

// ---------------------------------------------------------------------------
// Attention: one wave per (b, h, 16-query block). Two-pass flash softmax.
// Emits WMMA for QK^T and P*V; atomically accumulates per-key attention mass
// (mean over heads, summed over queries) into col[] for the gate.
// ---------------------------------------------------------------------------
__global__ __launch_bounds__(32) void attn_kernel(
    const float* __restrict__ qkv, const unsigned char* __restrict__ pad,
    float* __restrict__ ctx, float* __restrict__ col) {
  __shared__ __align__(16) float Pl[16][16];

  const int blk = blockIdx.x;
  const int qt  = blk & 127;
  const int h   = (blk >> 7) & 7;
  const int b   = blk >> 10;
  const int lane = threadIdx.x & 31;
  const int half = lane >> 4;
  const int l16  = lane & 15;

  const size_t rs = 3 * Dd;  // row stride of qkv
  const float* Qb = qkv + (size_t)(b * Sd) * rs + h * HD;
  const float* Kb = Qb + Dd;
  const float* Vb = Qb + 2 * Dd;
  const int q0 = qt * 16;
  const float scale = 0.125f;  // 1/sqrt(64)

  // Q fragments for all 16 K-steps (K dim = 64)
  v2f a[16];
#pragma unroll
  for (int t = 0; t < 16; ++t) {
    const float* p = Qb + (size_t)(q0 + l16) * rs + 4 * t + 2 * half;
    a[t].x = p[0]; a[t].y = p[1];
  }

  float runmax[8], runsum[8];
#pragma unroll
  for (int r = 0; r < 8; ++r) { runmax[r] = -3.0e38f; runsum[r] = 0.0f; }

  // ---- pass 1: per-row max and exp-sum ----
  for (int kt = 0; kt < Sd / 16; ++kt) {
    const int key0 = kt * 16;
    v8f s = {};
#pragma unroll
    for (int t = 0; t < 16; ++t) {
      v2f bf;
      const float* p = Kb + (size_t)(key0 + l16) * rs + 4 * t + 2 * half;
      bf.x = p[0]; bf.y = p[1];
      s = wmma_f32(a[t], bf, s);
    }
    const bool kp = pad[b * Sd + key0 + l16] != 0;
#pragma unroll
    for (int r = 0; r < 8; ++r) {
      float sc = kp ? -3.0e38f : s[r] * scale;
      float mx = sc;
      mx = fmaxf(mx, __shfl_xor(mx, 1, 32));
      mx = fmaxf(mx, __shfl_xor(mx, 2, 32));
      mx = fmaxf(mx, __shfl_xor(mx, 4, 32));
      mx = fmaxf(mx, __shfl_xor(mx, 8, 32));
      const float nm = fmaxf(runmax[r], mx);
      float e = kp ? 0.0f : expf(sc - nm);
      e += __shfl_xor(e, 1, 32);
      e += __shfl_xor(e, 2, 32);
      e += __shfl_xor(e, 4, 32);
      e += __shfl_xor(e, 8, 32);
      runsum[r] = runsum[r] * expf(runmax[r] - nm) + e;
      runmax[r] = nm;
    }
  }
  float invl[8];
#pragma unroll
  for (int r = 0; r < 8; ++r) invl[r] = 1.0f / runsum[r];

  // ---- pass 2: exact P, column sums, ctx = P * V ----
  v8f cacc[4] = {};
  for (int kt = 0; kt < Sd / 16; ++kt) {
    const int key0 = kt * 16;
    v8f s = {};
#pragma unroll
    for (int t = 0; t < 16; ++t) {
      v2f bf;
      const float* p = Kb + (size_t)(key0 + l16) * rs + 4 * t + 2 * half;
      bf.x = p[0]; bf.y = p[1];
      s = wmma_f32(a[t], bf, s);
    }
    const bool kp = pad[b * Sd + key0 + l16] != 0;
    float csum = 0.0f;
#pragma unroll
    for (int r = 0; r < 8; ++r) {
      const float pe = kp ? 0.0f : expf(s[r] * scale - runmax[r]) * invl[r];
      Pl[r + 8 * half][l16] = pe;  // C-layout -> (row, col) in LDS
      csum += pe;
    }
    // gate accumulation: sum over this block's 16 query rows, /H over heads
    csum += __shfl_xor(csum, 16, 32);
    if (half == 0) atomicAdd(&col[b * Sd + key0 + l16], csum * (1.0f / Hh));
    __syncthreads();  // single-wave WG: orders LDS write->read
#pragma unroll
    for (int t = 0; t < 4; ++t) {
      const int kk = 4 * t + 2 * half;
      v2f ap;
      ap.x = Pl[l16][kk]; ap.y = Pl[l16][kk + 1];  // A-fragment of P
#pragma unroll
      for (int nt = 0; nt < 4; ++nt) {
        v2f bv;
        const float* p = Vb + (size_t)(key0 + kk) * rs + nt * 16 + l16;
        bv.x = p[0];
        bv.y = p[rs];  // next key row
        cacc[nt] = wmma_f32(ap, bv, cacc[nt]);
      }
    }
    __syncthreads();
  }

  // store ctx [B,S,D], columns h*64..h*64+63
#pragma unroll
  for (int r = 0; r < 8; ++r) {
    const int m = q0 + r + 8 * half;
    float* out = ctx + (size_t)(b * Sd + m) * Dd + h * HD;
#pragma unroll
    for (int nt = 0; nt < 4; ++nt) out[nt * 16 + l16] = cacc[nt][r];
  }
}

// ---------------------------------------------------------------------------
// Gate + stable compaction: block scan over S=2048 flags per batch.
// ---------------------------------------------------------------------------
__global__ __launch_bounds__(1024) void gate_scan(
    const float* __restrict__ col, int* __restrict__ idxmap,
    int* __restrict__ selnum, float* __restrict__ ypad) {
  const int b = blockIdx.x, tid = threadIdx.x;
  const int k0 = tid * 2, k1 = tid * 2 + 1;
  const int f0 = (col[b * Sd + k0] >= 1.0f) ? 1 : 0;
  const int f1 = (col[b * Sd + k1] >= 1.0f) ? 1 : 0;
  __shared__ int sc[1024];
  sc[tid] = f0 + f1;
  __syncthreads();
  for (int off = 1; off < 1024; off <<= 1) {
    int v = (tid >= off) ? sc[tid - off] : 0;
    __syncthreads();
    sc[tid] += v;
    __syncthreads();
  }
  const int excl = sc[tid] - (f0 + f1);
  if (f0) idxmap[b * Sd + excl] = k0;
  if (f1) idxmap[b * Sd + excl + f0] = k1;
  const int total = sc[1023];
  if (tid == 0) selnum[b] = total;
  ypad[b * Sd + k0] = (k0 < total) ? 0.0f : 1.0f;
  ypad[b * Sd + k1] = (k1 < total) ? 0.0f : 1.0f;
}

// ---------------------------------------------------------------------------
// Gather selected rows, add, LayerNorm over D=512. One block (256 thr) / row.
// ---------------------------------------------------------------------------
__global__ __launch_bounds__(256) void gather_add_ln(
    const float* __restrict__ xattn, const float* __restrict__ src,
    const int* __restrict__ idxmap, const int* __restrict__ selnum,
    const float* __restrict__ g, const float* __restrict__ bta,
    float* __restrict__ out) {
  const int row = blockIdx.x;          // 0..B*S-1
  const int b = row / Sd, jj = row % Sd;
  const int tid = threadIdx.x;
  float v0 = 0.0f, v1 = 0.0f;
  const int sel = selnum[b];
  if (jj < sel) {
    const int srow = idxmap[b * Sd + jj];
    const size_t base = (size_t)(b * Sd + srow) * Dd;
    v0 = xattn[base + tid] + src[base + tid];
    v1 = xattn[base + tid + 256] + src[base + tid + 256];
  }
  __shared__ float red[256];
  red[tid] = v0 + v1;
  __syncthreads();
  for (int o = 128; o > 0; o >>= 1) { if (tid < o) red[tid] += red[tid + o]; __syncthreads(); }
  const float mean = red[0] * (1.0f / Dd);
  __syncthreads();
  const float d0 = v0 - mean, d1 = v1 - mean;
  red[tid] = d0 * d0 + d1 * d1;
  __syncthreads();
  for (int o = 128; o > 0; o >>= 1) { if (tid < o) red[tid] += red[tid + o]; __syncthreads(); }
  const float rstd = rsqrtf(red[0] * (1.0f / Dd) + LN_EPS);
  const size_t ob = (size_t)row * Dd;
  out[ob + tid]       = d0 * rstd * g[tid] + bta[tid];
  out[ob + tid + 256] = d1 * rstd * g[tid + 256] + bta[tid + 256];
}

__global__ __launch_bounds__(256) void add_ln(
    const float* __restrict__ x, const float* __restrict__ y,
    const float* __restrict__ g, const float* __restrict__ bta,
    float* __restrict__ out) {
  const int row = blockIdx.x;
  const int tid = threadIdx.x;
  const size_t base = (size_t)row * Dd;
  const float v0 = x[base + tid] + y[base + tid];
  const float v1 = x[base + tid + 256] + y[base + tid + 256];
  __shared__ float red[256];
  red[tid] = v0 + v1;
  __syncthreads();
  for (int o = 128; o > 0; o >>= 1) { if (tid < o) red[tid] += red[tid + o]; __syncthreads(); }
  const float mean = red[0] * (1.0f / Dd);
  __syncthreads();
  const float d0 = v0 - mean, d1 = v1 - mean;
  red[tid] = d0 * d0 + d1 * d1;
  __syncthreads();
  for (int o = 128; o > 0; o >>= 1) { if (tid < o) red[tid] += red[tid + o]; __syncthreads(); }
  const float rstd = rsqrtf(red[0] * (1.0f / Dd) + LN_EPS);
  out[base + tid]       = d0 * rstd * g[tid] + bta[tid];
  out[base + tid + 256] = d1 * rstd * g[tid + 256] + bta[tid + 256];
}

__global__ void zero_f32(float* __restrict__ p, int n) {
  const int i = blockIdx.x * 256 + threadIdx.x;
  if (i < n) p[i] = 0.0f;
}

// ---------------------------------------------------------------------------
extern "C" void kernel_launch(void* const* d_in, const int* in_sizes, int n_in,
                              void* d_out, int out_size, void* d_ws, size_t ws_size,
                              hipStream_t stream) {
  (void)in_sizes; (void)n_in; (void)out_size; (void)ws_size;
  const float* src            = (const float*)d_in[0];
  const unsigned char* src_pad = (const unsigned char*)d_in[1];
  const float* in_proj_w = (const float*)d_in[2];
  const float* in_proj_b = (const float*)d_in[3];
  const float* out_w     = (const float*)d_in[4];
  const float* out_b     = (const float*)d_in[5];
  const float* ln1_g     = (const float*)d_in[6];
  const float* ln1_b     = (const float*)d_in[7];
  const float* ln2_g     = (const float*)d_in[8];
  const float* ln2_b     = (const float*)d_in[9];
  const float* w1        = (const float*)d_in[10];
  const float* b1        = (const float*)d_in[11];
  const float* w2        = (const float*)d_in[12];
  const float* b2        = (const float*)d_in[13];

  float* out_y   = (float*)d_out;
  float* out_pad = out_y + (size_t)Bd * Sd * Dd;

  const int M = Bd * Sd;  // 4096
  float* ws = (float*)d_ws;
  float* qkv   = ws; ws += (size_t)M * 3 * Dd;   // [M, 1536]
  float* ctxb  = ws; ws += (size_t)M * Dd;       // [M, 512]
  float* xattn = ws; ws += (size_t)M * Dd;       // [M, 512]
  float* colb  = ws; ws += M;                    // [B*S]
  int*   idxmap = (int*)ws; ws += M;             // [B*S]
  int*   selnum = (int*)ws; ws += 8;             // [B] (padded, keeps 16B align)
  float* hbuf  = ws; ws += (size_t)M * Dd;       // [M, 512]
  float* hmid  = ws; ws += (size_t)M * FCd;      // [M, 2048]
  float* ffb   = ws;                             // [M, 512]

  // 1) zero gate accumulator
  zero_f32<<<(M + 255) / 256, 256, 0, stream>>>(colb, M);
  // 2) QKV projection: [M,512] x [1536,512]^T
  gemm_wmma_f32<0><<<dim3(3 * Dd / 64, M / 64), 256, 0, stream>>>(
      src, in_proj_w, in_proj_b, qkv, M, 3 * Dd, Dd);
  // 3) attention (+ gate column sums)
  attn_kernel<<<Bd * Hh * (Sd / 16), 32, 0, stream>>>(qkv, src_pad, ctxb, colb);
  // 4) output projection: [M,512] x [512,512]^T
  gemm_wmma_f32<0><<<dim3(Dd / 64, M / 64), 256, 0, stream>>>(
      ctxb, out_w, out_b, xattn, M, Dd, Dd);
  // 5) gate + compaction map + y_pad
  gate_scan<<<Bd, 1024, 0, stream>>>(colb, idxmap, selnum, out_pad);
  // 6) h = LN1(x_c + src_c)
  gather_add_ln<<<M, 256, 0, stream>>>(xattn, src, idxmap, selnum, ln1_g, ln1_b, hbuf);
  // 7) FFN1 + SiLU: [M,512] x [2048,512]^T
  gemm_wmma_f32<1><<<dim3(FCd / 64, M / 64), 256, 0, stream>>>(
      hbuf, w1, b1, hmid, M, FCd, Dd);
  // 8) FFN2: [M,2048] x [512,2048]^T
  gemm_wmma_f32<0><<<dim3(Dd / 64, M / 64), 256, 0, stream>>>(
      hmid, w2, b2, ffb, M, Dd, FCd);
  // 9) y = LN2(h + ff)
  add_ln<<<M, 256, 0, stream>>>(hbuf, ffb, ln2_g, ln2_b, out_y);
}